// pgeoNet_55387898250020
// MI455X (gfx1250) — compile-verified
//
#include <hip/hip_runtime.h>
#include <math.h>

typedef __attribute__((ext_vector_type(2))) float v2f;
typedef __attribute__((ext_vector_type(8))) float v8f;

#define KDEG 8
#define TWO_PI_F 6.28318530717958647692f

__device__ __forceinline__ float3 ldpos(const float* __restrict__ p, int n) {
    return make_float3(p[3 * n + 0], p[3 * n + 1], p[3 * n + 2]);
}
__device__ __forceinline__ float3 f3sub(float3 a, float3 b) {
    return make_float3(a.x - b.x, a.y - b.y, a.z - b.z);
}
__device__ __forceinline__ float3 f3cross(float3 a, float3 b) {
    return make_float3(a.y * b.z - a.z * b.y,
                       a.z * b.x - a.x * b.z,
                       a.x * b.y - a.y * b.x);
}
__device__ __forceinline__ float f3dot(float3 a, float3 b) {
    return a.x * b.x + a.y * b.y + a.z * b.z;
}

// One wave32 handles TWO edges. Lanes (row16 = lane&15) are the 16 A-matrix
// rows (edge = row16>>3, a = row16&7); half = lane>>4 selects the K=0,1 vs
// K=2,3 column pair of the 16x4 f32 A operand per the CDNA5 WMMA layout.
// a2 = P*P^T and b2 = P*W^T are each one V_WMMA_F32_16X16X4_F32; the two
// 8x8 diagonal blocks of the 16x16 D tile are the per-edge pairwise grids.
__global__ void __launch_bounds__(256)
triplet_wmma_kernel(const float* __restrict__ pos,
                    const int* __restrict__ ei,
                    float* __restrict__ out,
                    int* __restrict__ outI,
                    int E, int NQ) {
    const int T = E * KDEG;
    const int off_angle = E;
    const int off_t0 = E + T;
    const int off_t1 = E + 2 * T;
    const int off_t2 = E + 3 * T;
    const int off_ps = E + 4 * T;
    const int off_ji = E + 5 * T + NQ;
    const int off_kj = E + 6 * T + NQ;

    const int tid = blockIdx.x * blockDim.x + threadIdx.x;
    const int wave = tid >> 5;
    const int lane = tid & 31;
    const int row16 = lane & 15;
    const int half = lane >> 4;
    const int edge_sel = row16 >> 3;   // which of the 2 edges this row/col belongs to
    const int a_row = row16 & 7;       // neighbor slot within the edge

    int e = wave * 2 + edge_sel;
    if (e >= E) e = E - 1;             // clamp (not return): EXEC must stay all-1s for WMMA

    // ---- per-lane edge/row geometry (all in registers) ----
    const int j = ei[e];
    const int i = ei[E + e];
    const float3 pj = ldpos(pos, j);
    const float3 pi = ldpos(pos, i);
    const float3 u = f3sub(pi, pj);                  // pos_ji
    const float dji = sqrtf(f3dot(u, u));

    const int kjEdge = j * KDEG + a_row;             // edge id k->j
    const int k = ei[kjEdge];
    const int tgt = ei[E + kjEdge];
    __builtin_prefetch(&pos[3 * tgt], 0, 0);         // global_prefetch_b8
    const float3 pk = ldpos(pos, k);
    const float3 pjk = f3sub(pk, pj);
    const float3 pt = ldpos(pos, tgt);
    const float3 rel_kj = f3sub(pt, pk);
    const float dist_kj = sqrtf(f3dot(rel_kj, rel_kj));

    const float3 p = f3cross(u, pjk);                // plane2 row for slot a
    const float rdji = 1.0f / dji;
    float3 wv = f3cross(p, u);                       // b2[a][n] = p_a . (p_n x u)/|u|
    wv.x *= rdji; wv.y *= rdji; wv.z *= rdji;

    // ---- pack WMMA operands (16x4 f32 A layout; B mirrors it) ----
    v2f aop, bop;
    aop.x = half ? p.z : p.x;
    aop.y = half ? 0.0f : p.y;
    bop.x = half ? wv.z : wv.x;
    bop.y = half ? 0.0f : wv.y;

    const v8f zero = {};
    v8f a2 = __builtin_amdgcn_wmma_f32_16x16x4_f32(
        false, aop, false, aop, (short)0, zero, false, false);   // P * P^T
    v8f b2 = __builtin_amdgcn_wmma_f32_16x16x4_f32(
        false, aop, false, bop, (short)0, zero, false, false);   // P * W^T

    // ---- broadcast the 8 neighbor node ids of this lane's edge ----
    int ka[KDEG];
#pragma unroll
    for (int r = 0; r < KDEG; ++r)
        ka[r] = __shfl(k, edge_sel * KDEG + r, 32);

    // ---- torsion per (row r, this lane's column), mask, wrap ----
    float tors[KDEG];
#pragma unroll
    for (int r = 0; r < KDEG; ++r) {
        float t = atan2f(b2[r], a2[r]);
        t = (t <= 0.0f) ? t + TWO_PI_F : t;
        const bool valid = (k != i) && (k != ka[r]);
        tors[r] = valid ? t : __builtin_inff();
    }
    // min over the 8 columns of this edge: xor-1/2/4 stays inside 8-lane groups
#pragma unroll
    for (int m = 1; m <= 4; m <<= 1) {
#pragma unroll
        for (int r = 0; r < KDEG; ++r) {
            const float o = __shfl_xor(tors[r], m, 32);
            tors[r] = fminf(tors[r], o);
        }
    }
    float tmin = tors[0];
#pragma unroll
    for (int r = 1; r < KDEG; ++r)
        tmin = (a_row == r) ? tors[r] : tmin;

    const int t_idx = e * KDEG + a_row;

    // useful D lanes: column's edge matches the row half (diagonal 8x8 block)
    if (half == edge_sel) {
        out[off_t0 + t_idx] = tmin;
        out[off_t1 + t_idx] = tmin;
        out[off_t2 + t_idx] = tmin;
    }
    // A-row holder lanes (0..15) emit angle / plane_s / idx / dist
    if (half == 0) {
        const float dotv = f3dot(u, pjk);
        const float bnorm = sqrtf(f3dot(p, p));
        const float ang = atan2f(bnorm, dotv);
        out[off_angle + t_idx] = ang;
        out[off_ps + t_idx] = 0.5f * sinf(ang) * dist_kj * dji;
        outI[off_ji + t_idx] = e;
        outI[off_kj + t_idx] = kjEdge;
        if (a_row == 0) out[e] = dji;   // dist region at offset 0
    }
}

__global__ void __launch_bounds__(256)
volume_kernel(const float* __restrict__ pos,
              const int* __restrict__ qi,
              float* __restrict__ out,
              int E, int NQ) {
    const int q = blockIdx.x * blockDim.x + threadIdx.x;
    if (q >= NQ) return;
    const int i0 = qi[q];
    const int i1 = qi[NQ + q];
    const int i2 = qi[2 * NQ + q];
    const int i3 = qi[3 * NQ + q];
    const float3 p0 = ldpos(pos, i0);
    const float3 v1 = f3sub(ldpos(pos, i1), p0);
    const float3 v2 = f3sub(ldpos(pos, i2), p0);
    const float3 v3 = f3sub(ldpos(pos, i3), p0);
    const float vol = fabsf(f3dot(f3cross(v1, v2), v3)) * (1.0f / 6.0f);
    const int off_vol = E + 5 * (E * KDEG);
    out[off_vol + q] = vol;
}

extern "C" void kernel_launch(void* const* d_in, const int* in_sizes, int n_in,
                              void* d_out, int out_size, void* d_ws, size_t ws_size,
                              hipStream_t stream) {
    const float* pos = (const float*)d_in[0];
    const int* ei = (const int*)d_in[1];
    const int* qi = (const int*)d_in[2];
    const int E = in_sizes[1] / 2;
    const int NQ = in_sizes[2] / 4;
    float* out = (float*)d_out;
    int* outI = (int*)d_out;

    // 2 edges per wave32 -> ceil(E/2) waves
    const long long waves = (E + 1) / 2;
    const long long threads = waves * 32;
    const int block = 256;
    const int grid = (int)((threads + block - 1) / block);
    triplet_wmma_kernel<<<grid, block, 0, stream>>>(pos, ei, out, outI, E, NQ);

    const int vgrid = (NQ + 255) / 256;
    volume_kernel<<<vgrid, 256, 0, stream>>>(pos, qi, out, E, NQ);
}